// ModelMixPromptDID_29248727286321
// MI455X (gfx1250) — compile-verified
//
#include <hip/hip_runtime.h>
#include <hip/hip_bf16.h>
#include <stdint.h>

// ---------------------------------------------------------------------------
// Cross-entropy over V=32000 vocab, B*T=4096 rows. Memory-bound: 512 MB of
// fp32 logits -> ~22us floor at 23.3 TB/s. Strategy: TDM (tensor_load_to_lds)
// DMAs each 128KB logits row into LDS (two half-row descriptors so compute on
// half 0 overlaps the DMA of half 1); both softmax passes then run from LDS.
// ---------------------------------------------------------------------------

typedef __attribute__((ext_vector_type(4))) unsigned int u32x4;
typedef __attribute__((ext_vector_type(8))) int          i32x8;
typedef __attribute__((ext_vector_type(4))) int          i32x4;

#define CE_EPS    1e-8f
#define THREADS   256
#define NWAVES    (THREADS / 32)

__device__ __forceinline__ float wave_max_f32(float v) {
    #pragma unroll
    for (int off = 16; off > 0; off >>= 1)
        v = fmaxf(v, __shfl_xor(v, off, 32));
    return v;
}
__device__ __forceinline__ float wave_sum_f32(float v) {
    #pragma unroll
    for (int off = 16; off > 0; off >>= 1)
        v += __shfl_xor(v, off, 32);
    return v;
}

// Issue one TDM load of `nelem` fp32 from gaddr -> LDS byte addr lds_addr.
__device__ __forceinline__ void tdm_load_row(uint64_t gaddr, uint32_t lds_addr, int nelem) {
    u32x4 g0;
    g0.x = 1u;                                            // count=1 (valid D#)
    g0.y = lds_addr;                                      // lds_addr
    g0.z = (uint32_t)(gaddr & 0xffffffffu);               // global_addr[31:0]
    g0.w = (uint32_t)((gaddr >> 32) & 0x01ffffffu)        // global_addr[56:32]
         | (2u << 30);                                    // type = 2 ("image")
    i32x8 g1;
    g1[0] = 0x00020000;                                   // data_size = 2 (4 bytes)
    g1[1] = (int)((nelem & 0xffff) << 16);                // tensor_dim0[15:0]
    g1[2] = (int)(((unsigned)nelem >> 16) | (1u << 16));  // tensor_dim0[31:16] | tensor_dim1=1
    g1[3] = (int)((nelem & 0xffff) << 16);                // tile_dim0 = nelem (16-bit field)
    g1[4] = 1;                                            // tile_dim1 = 1, tile_dim2 = 0
    g1[5] = nelem;                                        // tensor_dim0_stride[31:0]
    g1[6] = 0;
    g1[7] = 0;
    i32x4 z4 = {0, 0, 0, 0};
    i32x8 z8 = {0, 0, 0, 0, 0, 0, 0, 0};
    __builtin_amdgcn_tensor_load_to_lds(g0, g1, z4, z4, z8, 0);
}

extern __shared__ __align__(16) float smem[];   // V floats (128 KB)

__global__ __launch_bounds__(THREADS)
void ce_row_kernel(const float* __restrict__ logits,
                   const int*   __restrict__ labels,
                   const int*   __restrict__ ents_p,
                   float*       __restrict__ row_loss,
                   int T, int V)
{
    __shared__ float red[NWAVES];

    const int r    = blockIdx.x;           // row id in [0, B*T)
    const int ENTS = *ents_p;              // uniform scalar load
    const int b    = r / T;
    const int t    = r - b * T;
    const int H    = V >> 1;               // half row (V even)

    // byte address of logits[b, ENTS + t, 0]
    const uint64_t row_idx = (uint64_t)b * (uint64_t)(ENTS + T) + (uint64_t)(ENTS + t);
    const uint64_t gaddr   = (uint64_t)(uintptr_t)logits + row_idx * (uint64_t)V * 4ull;
    const uint32_t lds_base = (uint32_t)(uintptr_t)smem;   // low 32 bits = LDS byte addr

    const int tid  = threadIdx.x;
    const int wave = tid >> 5;

    // ---- Wave 0 queues two TDM half-row loads (in-order on TENSORcnt) ----
    if (wave == 0) {
        tdm_load_row(gaddr,                    lds_base,                    H);
        tdm_load_row(gaddr + (uint64_t)H * 4u, lds_base + (uint32_t)H * 4u, H);
    }
    __builtin_amdgcn_s_wait_tensorcnt(1);   // half 0 landed (no-op on waves 1..7)
    __syncthreads();

    // ---- Pass 1a: max over half 0 while half 1 still streams ----
    float m = -INFINITY;
    for (int i = tid * 4; i < H; i += THREADS * 4) {
        float4 v = *(const float4*)(&smem[i]);
        m = fmaxf(m, fmaxf(fmaxf(v.x, v.y), fmaxf(v.z, v.w)));
    }
    __builtin_amdgcn_s_wait_tensorcnt(0);   // half 1 landed
    __syncthreads();

    // ---- Pass 1b: max over half 1 ----
    for (int i = H + tid * 4; i < V; i += THREADS * 4) {
        float4 v = *(const float4*)(&smem[i]);
        m = fmaxf(m, fmaxf(fmaxf(v.x, v.y), fmaxf(v.z, v.w)));
    }
    m = wave_max_f32(m);
    if ((tid & 31) == 0) red[wave] = m;
    __syncthreads();
    if (wave == 0) {
        float mm = (tid < NWAVES) ? red[tid] : -INFINITY;
        mm = wave_max_f32(mm);
        if (tid == 0) red[0] = mm;
    }
    __syncthreads();
    m = red[0];
    __syncthreads();                        // red[0] consumed before reuse

    // ---- Pass 2: sum of exp(x - m) over the full row (all in LDS) ----
    float s = 0.0f;
    for (int i = tid * 4; i < V; i += THREADS * 4) {
        float4 v = *(const float4*)(&smem[i]);
        s += __expf(v.x - m) + __expf(v.y - m) + __expf(v.z - m) + __expf(v.w - m);
    }
    s = wave_sum_f32(s);
    if ((tid & 31) == 0) red[wave] = s;
    __syncthreads();

    if (tid == 0) {
        float tot = 0.0f;
        #pragma unroll
        for (int w = 0; w < NWAVES; ++w) tot += red[w];
        const int   lab = labels[r];
        const float xl  = smem[lab];
        const float p   = __expf(xl - m) / tot;
        row_loss[r]     = -logf(p + CE_EPS);
    }
}

__global__ __launch_bounds__(THREADS)
void ce_reduce_kernel(const float* __restrict__ row_loss,
                      float*       __restrict__ out, int n)
{
    __shared__ float red[THREADS];
    float s = 0.0f;
    for (int i = threadIdx.x; i < n; i += THREADS) s += row_loss[i];
    red[threadIdx.x] = s;
    __syncthreads();
    #pragma unroll
    for (int off = THREADS / 2; off > 0; off >>= 1) {
        if (threadIdx.x < off) red[threadIdx.x] += red[threadIdx.x + off];
        __syncthreads();
    }
    if (threadIdx.x == 0) out[0] = red[0] / (float)n;
}

extern "C" void kernel_launch(void* const* d_in, const int* in_sizes, int n_in,
                              void* d_out, int out_size, void* d_ws, size_t ws_size,
                              hipStream_t stream)
{
    const float* logits = (const float*)d_in[0];
    const int*   labels = (const int*)d_in[1];
    const int*   ents   = (const int*)d_in[2];
    float*       out    = (float*)d_out;
    float*       rowls  = (float*)d_ws;     // BT floats of scratch

    const int BT = in_sizes[1];             // B*T = 4096
    const int T  = 512;                     // reference shapes
    const int V  = 32000;

    const size_t shmem = (size_t)V * sizeof(float);   // 128 KB (LDS is 320 KB/WGP)
    (void)hipFuncSetAttribute((const void*)ce_row_kernel,
                              hipFuncAttributeMaxDynamicSharedMemorySize, (int)shmem);

    ce_row_kernel<<<BT, THREADS, shmem, stream>>>(logits, labels, ents, rowls, T, V);
    ce_reduce_kernel<<<1, THREADS, 0, stream>>>(rowls, out, BT);
}